// GATNet_20581483283117
// MI455X (gfx1250) — compile-verified
//
#include <hip/hip_runtime.h>
#include <math.h>

// ---------------------------------------------------------------------------
// GATNet forward for MI455X (gfx1250, wave32).
// GEMMs use V_WMMA_F32_16X16X4_F32 (exact fp32, matches reference math).
// Scatter softmax / aggregation via native f32/u32 global atomics.
// ---------------------------------------------------------------------------

typedef __attribute__((ext_vector_type(2))) float v2f;
typedef __attribute__((ext_vector_type(8))) float v8f;

constexpr int KC   = 64;       // K-chunk staged in LDS
constexpr int LDSW = KC + 2;   // pad to 66 floats -> conflict-free B reads

// Monotonic float <-> uint mapping for atomicMax on floats.
__device__ __forceinline__ unsigned fkey(float f) {
  unsigned u = __float_as_uint(f);
  return (u & 0x80000000u) ? ~u : (u | 0x80000000u);
}
__device__ __forceinline__ float fdecode(unsigned v) {
  return (v & 0x80000000u) ? __uint_as_float(v & 0x7fffffffu)
                           : __uint_as_float(~v);
}

// ---------------------------------------------------------------------------
// Accumulate C[row0..row0+15, 0..127] += A[rows, K] * W[128, K]^T using
// v_wmma_f32_16x16x4_f32. One wave per 16-row strip; 8 waves per block stage
// the shared W chunk cooperatively. Optional per-row scale (masking).
//
// A-frag layout (32-bit A, 16x4): lanes 0-15 hold M=lane {K=k,k+1},
// lanes 16-31 hold M=lane-16 {K=k+2,k+3}.
// B-frag: lanes = N index (col of W^T), VGPRs = K pair, mirrored halves.
// C/D: VGPR v -> row v (lanes 0-15) / row v+8 (lanes 16-31), col = lane&15.
// ---------------------------------------------------------------------------
__device__ __forceinline__ void gemm_tile_accum(
    v8f acc[8], const float* __restrict__ A, const float* __restrict__ W,
    int K, int row0, int nrows, const float* __restrict__ rowScale,
    float* lds, int tid)
{
  const int lane = tid & 31;
  const int half = lane >> 4;
  const int lr   = lane & 15;
  const int rowA = row0 + lr;
  const bool rv  = rowA < nrows;
  const float sc = rv ? (rowScale ? rowScale[rowA] : 1.0f) : 0.0f;
  const float* Arow = A + (size_t)(rv ? rowA : 0) * K;

  for (int k0 = 0; k0 < K; k0 += KC) {
    __syncthreads();
    for (int i = tid; i < 128 * KC; i += (int)blockDim.x) {
      const int j  = i >> 6;          // 0..127 (output col / W row)
      const int kk = i & (KC - 1);    // 0..63
      lds[j * LDSW + kk] = W[(size_t)j * K + k0 + kk];
    }
    __syncthreads();

    if (k0 + KC < K)  // warm next A chunk (global_prefetch_b8)
      __builtin_prefetch(Arow + k0 + KC, 0, 1);

    for (int kk = 0; kk < KC; kk += 4) {
      const int ko = kk + half * 2;
      const float2 av = *(const float2*)(Arow + k0 + ko);
      v2f a; a.x = av.x * sc; a.y = av.y * sc;
      #pragma unroll
      for (int j = 0; j < 8; ++j) {
        const float2 bv = *(const float2*)(lds + (j * 16 + lr) * LDSW + ko);
        v2f b; b.x = bv.x; b.y = bv.y;
        acc[j] = __builtin_amdgcn_wmma_f32_16x16x4_f32(
            /*neg_a=*/false, a, /*neg_b=*/false, b,
            /*c_mod=*/(short)0, acc[j], /*reuse_a=*/false, /*reuse_b=*/false);
      }
    }
  }
}

// ---------------------------------------------------------------------------
// Fused input projection:
// h = prelu0( num_x*num_mask @ num_proj_w^T + num_proj_b
//           + txt_x*txt_mask @ txt_proj_w^T + txt_proj_b
//           + x @ node_proj_w^T + node_proj_b )
// ---------------------------------------------------------------------------
__global__ void k_input_proj(
    const float* __restrict__ x, const float* __restrict__ num_x,
    const float* __restrict__ num_mask, const float* __restrict__ txt_x,
    const float* __restrict__ txt_mask,
    const float* __restrict__ num_proj_w, const float* __restrict__ num_proj_b,
    const float* __restrict__ txt_proj_w, const float* __restrict__ txt_proj_b,
    const float* __restrict__ node_proj_w, const float* __restrict__ node_proj_b,
    const float* __restrict__ prelu0_a, float* __restrict__ h, int N)
{
  __shared__ float lds[128 * LDSW];
  const int tid  = threadIdx.x;
  const int wave = tid >> 5;
  const int row0 = blockIdx.x * 128 + wave * 16;

  v8f acc[8];
  v8f zero = {};
  #pragma unroll
  for (int j = 0; j < 8; ++j) acc[j] = zero;

  gemm_tile_accum(acc, txt_x, txt_proj_w, 384, row0, N, txt_mask, lds, tid);
  gemm_tile_accum(acc, x,     node_proj_w, 128, row0, N, nullptr,  lds, tid);

  const int lane = tid & 31, half = lane >> 4, lr = lane & 15;
  #pragma unroll
  for (int j = 0; j < 8; ++j) {
    const int col   = j * 16 + lr;
    const float npw = num_proj_w[col];
    const float cb  = num_proj_b[col] + txt_proj_b[col] + node_proj_b[col];
    const float pa  = prelu0_a[col];
    #pragma unroll
    for (int v = 0; v < 8; ++v) {
      const int row = row0 + v + half * 8;
      if (row < N) {
        float val = acc[j][v] + num_x[row] * num_mask[row] * npw + cb;
        val = (val >= 0.0f) ? val : pa * val;
        h[(size_t)row * 128 + col] = val;
      }
    }
  }
}

// xh = h @ W^T  (GATConv linear, no bias)
__global__ void k_conv_gemm(const float* __restrict__ h,
                            const float* __restrict__ W,
                            float* __restrict__ xh, int N)
{
  __shared__ float lds[128 * LDSW];
  const int tid  = threadIdx.x;
  const int wave = tid >> 5;
  const int row0 = blockIdx.x * 128 + wave * 16;

  v8f acc[8];
  v8f zero = {};
  #pragma unroll
  for (int j = 0; j < 8; ++j) acc[j] = zero;

  gemm_tile_accum(acc, h, W, 128, row0, N, nullptr, lds, tid);

  const int lane = tid & 31, half = lane >> 4, lr = lane & 15;
  #pragma unroll
  for (int j = 0; j < 8; ++j) {
    const int col = j * 16 + lr;
    #pragma unroll
    for (int v = 0; v < 8; ++v) {
      const int row = row0 + v + half * 8;
      if (row < N) xh[(size_t)row * 128 + col] = acc[j][v];
    }
  }
}

// a_src[n,h] = <xh[n,h,:], att_src[h,:]>, same for a_dst. One thread per (n,h).
__global__ void k_attn_prep(const float* __restrict__ xh,
                            const float* __restrict__ att_src,
                            const float* __restrict__ att_dst,
                            float* __restrict__ a_src, float* __restrict__ a_dst,
                            int N)
{
  const int idx = blockIdx.x * blockDim.x + threadIdx.x;
  if (idx >= N * 4) return;
  const int n = idx >> 2, hd = idx & 3;
  const float4* xv = (const float4*)(xh + (size_t)n * 128 + hd * 32);
  const float4* sv = (const float4*)(att_src + hd * 32);
  const float4* dv = (const float4*)(att_dst + hd * 32);
  float ss = 0.0f, sd = 0.0f;
  #pragma unroll
  for (int i = 0; i < 8; ++i) {
    const float4 v = xv[i], s = sv[i], d = dv[i];
    ss += v.x * s.x + v.y * s.y + v.z * s.z + v.w * s.w;
    sd += v.x * d.x + v.y * d.y + v.z * d.z + v.w * d.w;
  }
  a_src[idx] = ss;
  a_dst[idx] = sd;
}

__device__ __forceinline__ void edge_sd(const int* ei, int e, int E, int& s, int& d) {
  if (e < E) { s = ei[e]; d = ei[E + e]; }
  else       { s = d = e - E; }           // self-loops appended
}

// Pass 1: alpha = leaky_relu(a_src[s]+a_dst[d], 0.2); segment max over dst.
__global__ void k_edge_alpha(const int* __restrict__ ei,
                             const float* __restrict__ a_src,
                             const float* __restrict__ a_dst,
                             float* __restrict__ exb, unsigned* __restrict__ amax,
                             int E, int N)
{
  const int e = blockIdx.x * blockDim.x + threadIdx.x;
  if (e >= E + N) return;
  int s, d; edge_sd(ei, e, E, s, d);
  #pragma unroll
  for (int hd = 0; hd < 4; ++hd) {
    float al = a_src[(size_t)s * 4 + hd] + a_dst[(size_t)d * 4 + hd];
    al = (al > 0.0f) ? al : 0.2f * al;
    exb[(size_t)e * 4 + hd] = al;
    atomicMax(&amax[(size_t)d * 4 + hd], fkey(al));
  }
}

// Pass 2: ex = exp(alpha - amax[dst]); segment sum denominator.
__global__ void k_edge_exp(const int* __restrict__ ei,
                           float* __restrict__ exb,
                           const unsigned* __restrict__ amax,
                           float* __restrict__ denom, int E, int N)
{
  const int e = blockIdx.x * blockDim.x + threadIdx.x;
  if (e >= E + N) return;
  int s, d; edge_sd(ei, e, E, s, d);
  #pragma unroll
  for (int hd = 0; hd < 4; ++hd) {
    const float m  = fdecode(amax[(size_t)d * 4 + hd]);
    const float ex = __expf(exb[(size_t)e * 4 + hd] - m);
    exb[(size_t)e * 4 + hd] = ex;
    atomicAdd(&denom[(size_t)d * 4 + hd], ex);
  }
}

// Pass 3: agg[dst] += xh[src] * coef. One wave per edge, float4 per lane.
__global__ void k_edge_aggregate(const int* __restrict__ ei,
                                 const float* __restrict__ xh,
                                 const float* __restrict__ exb,
                                 const float* __restrict__ denom,
                                 float* __restrict__ agg, int E, int N)
{
  const long long gid = (long long)blockIdx.x * blockDim.x + threadIdx.x;
  const int e = (int)(gid >> 5);
  const int lane = threadIdx.x & 31;
  if (e >= E + N) return;
  int s, d; edge_sd(ei, e, E, s, d);
  const int hd = lane >> 3;                       // 4 floats/lane -> head = (4*lane)/32
  const float coef = exb[(size_t)e * 4 + hd] / denom[(size_t)d * 4 + hd];
  const float4 v = *(const float4*)(xh + (size_t)s * 128 + lane * 4);
  float* p = agg + (size_t)d * 128 + lane * 4;
  atomicAdd(p + 0, v.x * coef);
  atomicAdd(p + 1, v.y * coef);
  atomicAdd(p + 2, v.z * coef);
  atomicAdd(p + 3, v.w * coef);
}

// h = prelu(layernorm(agg + bias) * g + b). One wave per node (wave32).
__global__ void k_node_post(const float* __restrict__ agg,
                            const float* __restrict__ bias,
                            const float* __restrict__ g, const float* __restrict__ b,
                            const float* __restrict__ pa,
                            float* __restrict__ h, int N)
{
  const long long gid = (long long)blockIdx.x * blockDim.x + threadIdx.x;
  const int n = (int)(gid >> 5);
  const int lane = threadIdx.x & 31;
  if (n >= N) return;

  float4 v  = *(const float4*)(agg  + (size_t)n * 128 + lane * 4);
  const float4 bb = *(const float4*)(bias + lane * 4);
  v.x += bb.x; v.y += bb.y; v.z += bb.z; v.w += bb.w;

  float s = v.x + v.y + v.z + v.w;
  #pragma unroll
  for (int o = 16; o > 0; o >>= 1) s += __shfl_xor(s, o, 32);
  const float mean = s * (1.0f / 128.0f);

  const float dx = v.x - mean, dy = v.y - mean, dz = v.z - mean, dw = v.w - mean;
  float q = dx * dx + dy * dy + dz * dz + dw * dw;
  #pragma unroll
  for (int o = 16; o > 0; o >>= 1) q += __shfl_xor(q, o, 32);
  const float rstd = rsqrtf(q * (1.0f / 128.0f) + 1e-5f);

  const float4 gg = *(const float4*)(g  + lane * 4);
  const float4 nb = *(const float4*)(b  + lane * 4);
  const float4 pp = *(const float4*)(pa + lane * 4);
  float4 o4;
  o4.x = dx * rstd * gg.x + nb.x;  o4.x = (o4.x >= 0.0f) ? o4.x : pp.x * o4.x;
  o4.y = dy * rstd * gg.y + nb.y;  o4.y = (o4.y >= 0.0f) ? o4.y : pp.y * o4.y;
  o4.z = dz * rstd * gg.z + nb.z;  o4.z = (o4.z >= 0.0f) ? o4.z : pp.z * o4.z;
  o4.w = dw * rstd * gg.w + nb.w;  o4.w = (o4.w >= 0.0f) ? o4.w : pp.w * o4.w;
  *(float4*)(h + (size_t)n * 128 + lane * 4) = o4;
}

// y[n] = <h[n,:], out_w> + out_b. One wave per node.
__global__ void k_output(const float* __restrict__ h,
                         const float* __restrict__ out_w,
                         const float* __restrict__ out_b,
                         float* __restrict__ y, int N)
{
  const long long gid = (long long)blockIdx.x * blockDim.x + threadIdx.x;
  const int n = (int)(gid >> 5);
  const int lane = threadIdx.x & 31;
  if (n >= N) return;
  const float4 v = *(const float4*)(h + (size_t)n * 128 + lane * 4);
  const float4 w = *(const float4*)(out_w + lane * 4);
  float s = v.x * w.x + v.y * w.y + v.z * w.z + v.w * w.w;
  #pragma unroll
  for (int o = 16; o > 0; o >>= 1) s += __shfl_xor(s, o, 32);
  if (lane == 0) y[n] = s + out_b[0];
}

// Zero aggregation + softmax state each call (graph-replay safe).
__global__ void k_init(float* __restrict__ agg, float* __restrict__ denom,
                       unsigned* __restrict__ amax, int N)
{
  const int i = blockIdx.x * blockDim.x + threadIdx.x;
  if (i < N * 128) agg[i] = 0.0f;
  if (i < N * 4) { denom[i] = 0.0f; amax[i] = 0u; }  // key 0 < key(-inf)
}

// ---------------------------------------------------------------------------
extern "C" void kernel_launch(void* const* d_in, const int* in_sizes, int n_in,
                              void* d_out, int out_size, void* d_ws, size_t ws_size,
                              hipStream_t stream) {
  const float* x          = (const float*)d_in[0];
  const float* num_x      = (const float*)d_in[1];
  const float* num_mask   = (const float*)d_in[2];
  const float* txt_x      = (const float*)d_in[3];
  const float* txt_mask   = (const float*)d_in[4];
  const int*   ei         = (const int*)  d_in[5];
  const float* num_proj_w = (const float*)d_in[6];
  const float* num_proj_b = (const float*)d_in[7];
  const float* txt_proj_w = (const float*)d_in[8];
  const float* txt_proj_b = (const float*)d_in[9];
  const float* node_proj_w= (const float*)d_in[10];
  const float* node_proj_b= (const float*)d_in[11];
  const float* prelu0_a   = (const float*)d_in[12];
  const float* conv_w[2]  = {(const float*)d_in[13], (const float*)d_in[20]};
  const float* att_s[2]   = {(const float*)d_in[14], (const float*)d_in[21]};
  const float* att_d[2]   = {(const float*)d_in[15], (const float*)d_in[22]};
  const float* bias[2]    = {(const float*)d_in[16], (const float*)d_in[23]};
  const float* norm_g[2]  = {(const float*)d_in[17], (const float*)d_in[24]};
  const float* norm_b[2]  = {(const float*)d_in[18], (const float*)d_in[25]};
  const float* prelu_a[2] = {(const float*)d_in[19], (const float*)d_in[26]};
  const float* out_w      = (const float*)d_in[27];
  const float* out_b      = (const float*)d_in[28];

  const int N  = in_sizes[0] / 128;
  const int E  = in_sizes[5] / 2;
  const int Et = E + N;

  float* ws = (float*)d_ws;
  float* h     = ws;              ws += (size_t)N * 128;
  float* xh    = ws;              ws += (size_t)N * 128;
  float* agg   = ws;              ws += (size_t)N * 128;
  float* a_src = ws;              ws += (size_t)N * 4;
  float* a_dst = ws;              ws += (size_t)N * 4;
  unsigned* amax = (unsigned*)ws; ws += (size_t)N * 4;
  float* denom = ws;              ws += (size_t)N * 4;
  float* exb   = ws;              // Et*4 floats

  const dim3 blk(256);
  const int gemmGrid = (N + 127) / 128;
  const int nodeHeadGrid = (N * 4 + 255) / 256;
  const int edgeGrid = (Et + 255) / 256;
  const unsigned edgeWaveGrid = (unsigned)(((long long)Et * 32 + 255) / 256);
  const unsigned nodeWaveGrid = (unsigned)(((long long)N * 32 + 255) / 256);
  const int initGrid = (N * 128 + 255) / 256;

  k_input_proj<<<gemmGrid, blk, 0, stream>>>(
      x, num_x, num_mask, txt_x, txt_mask,
      num_proj_w, num_proj_b, txt_proj_w, txt_proj_b,
      node_proj_w, node_proj_b, prelu0_a, h, N);

  for (int L = 0; L < 2; ++L) {
    k_init<<<initGrid, blk, 0, stream>>>(agg, denom, amax, N);
    k_conv_gemm<<<gemmGrid, blk, 0, stream>>>(h, conv_w[L], xh, N);
    k_attn_prep<<<nodeHeadGrid, blk, 0, stream>>>(xh, att_s[L], att_d[L], a_src, a_dst, N);
    k_edge_alpha<<<edgeGrid, blk, 0, stream>>>(ei, a_src, a_dst, exb, amax, E, N);
    k_edge_exp<<<edgeGrid, blk, 0, stream>>>(ei, exb, amax, denom, E, N);
    k_edge_aggregate<<<edgeWaveGrid, blk, 0, stream>>>(ei, xh, exb, denom, agg, E, N);
    k_node_post<<<nodeWaveGrid, blk, 0, stream>>>(agg, bias[L], norm_g[L], norm_b[L],
                                                  prelu_a[L], h, N);
  }

  k_output<<<nodeWaveGrid, blk, 0, stream>>>(h, out_w, out_b, (float*)d_out, N);
}